// SparseBox3DDecoderLite_62517543960667
// MI455X (gfx1250) — compile-verified
//
#include <hip/hip_runtime.h>
#include <hip/hip_bf16.h>
#include <math.h>

// ---------------- problem constants (from reference setup_inputs) -----------
#define BATCH   32
#define QN      100000
#define CN      10
#define NELEM   (QN * CN)          // 1,000,000 scores per batch
#define KDET    300
#define NB      4096               // histogram buckets (top 12 bits of key)
#define CAND_MAX 8192              // power of two, >> expected ~700 candidates
#define PB      32                 // blocks per batch for streaming kernels
#define TILE_FLOATS 1024           // 256 threads * 16B async chunks = 4KB tile

// Order-preserving float -> uint key (sigmoid is monotonic, select on logits)
__device__ __forceinline__ unsigned mono_key(float x) {
    unsigned u = __float_as_uint(x);
    return (u & 0x80000000u) ? ~u : (u | 0x80000000u);
}
__device__ __forceinline__ float key_to_float(unsigned k) {
    unsigned u = (k & 0x80000000u) ? (k ^ 0x80000000u) : ~k;
    return __uint_as_float(u);
}

__device__ __forceinline__ void wait_asynccnt0() {
#if __has_builtin(__builtin_amdgcn_s_wait_asynccnt)
    __builtin_amdgcn_s_wait_asynccnt(0);
#else
    asm volatile("s_wait_asynccnt 0" ::: "memory");
#endif
}
__device__ __forceinline__ void wait_asynccnt1() {
#if __has_builtin(__builtin_amdgcn_s_wait_asynccnt)
    __builtin_amdgcn_s_wait_asynccnt(1);
#else
    asm volatile("s_wait_asynccnt 1" ::: "memory");
#endif
}

// Per-lane 16B async global->LDS copy: writes LDS address ldsOff,
// reads global address sbase + gOff. Tracked by ASYNCcnt (in-order completion).
__device__ __forceinline__ void async_copy16(unsigned ldsOff, unsigned gOff,
                                             unsigned long long sbase) {
    asm volatile("global_load_async_to_lds_b128 %0, %1, %2"
                 :: "v"(ldsOff), "v"(gOff), "s"(sbase) : "memory");
}

// ---------------- kernel 0: zero hist + counters ----------------------------
__global__ void init_kernel(unsigned* __restrict__ ws_u32, int n) {
    int i = blockIdx.x * blockDim.x + threadIdx.x;
    if (i < n) ws_u32[i] = 0u;
}

// ---------------- kernel 1: per-batch 4096-bucket histogram -----------------
// Double-buffered async global->LDS pipeline: issue tile t+1, wait asynccnt<=1
// (tile t complete, t+1 still in flight), histogram tile t from LDS.
// Default RT temporal hints keep cls_scores resident in the 192MB L2 so the
// collect pass reads from L2 instead of HBM.
__global__ __launch_bounds__(256) void hist_kernel(const float* __restrict__ cls,
                                                   unsigned* __restrict__ hist) {
    __shared__ unsigned lh[NB];                 // 16 KB
    __shared__ float    buf[2 * TILE_FLOATS];   // 2 x 4 KB staging tiles
    const int tid = threadIdx.x;
    const int b   = blockIdx.x / PB;
    const int blk = blockIdx.x % PB;

    for (int i = tid; i < NB; i += 256) lh[i] = 0u;
    __syncthreads();

    const float* base = cls + (size_t)b * NELEM;
    const unsigned long long sbase = (unsigned long long)(size_t)base;
    const unsigned ldsBase = (unsigned)(size_t)(&buf[0]);
    const int numTiles = NELEM / TILE_FLOATS;   // 976 full tiles

    int cur = 0;
    // prologue: issue first tile
    if (blk < numTiles)
        async_copy16(ldsBase + (unsigned)(tid * 16),
                     (unsigned)(blk * TILE_FLOATS * 4 + tid * 16), sbase);

    for (int t = blk; t < numTiles; t += PB) {
        const int tn = t + PB;
        const bool hasNext = (tn < numTiles);
        if (hasNext)
            async_copy16(ldsBase + (unsigned)((1 - cur) * TILE_FLOATS * 4 + tid * 16),
                         (unsigned)(tn * TILE_FLOATS * 4 + tid * 16), sbase);
        // in-order completion: <=1 outstanding means tile t has landed in LDS
        if (hasNext) wait_asynccnt1(); else wait_asynccnt0();
        __syncthreads();

        float4 v = *reinterpret_cast<const float4*>(&buf[cur * TILE_FLOATS + tid * 4]);
        atomicAdd(&lh[mono_key(v.x) >> 20], 1u);
        atomicAdd(&lh[mono_key(v.y) >> 20], 1u);
        atomicAdd(&lh[mono_key(v.z) >> 20], 1u);
        atomicAdd(&lh[mono_key(v.w) >> 20], 1u);
        __syncthreads();   // everyone done reading buf[cur] before it is refilled
        cur ^= 1;
    }
    // tail (NELEM % TILE_FLOATS = 576 elements), handled by one block per batch
    if (blk == 0) {
        for (int i = numTiles * TILE_FLOATS + tid; i < NELEM; i += 256)
            atomicAdd(&lh[mono_key(base[i]) >> 20], 1u);
    }
    __syncthreads();

    unsigned* gh = hist + (size_t)b * NB;
    for (int i = tid; i < NB; i += 256)
        if (lh[i]) atomicAdd(&gh[i], lh[i]);
}

// ---------------- kernel 2: find threshold bucket per batch -----------------
__global__ void thresh_kernel(const unsigned* __restrict__ hist,
                              unsigned* __restrict__ threshBucket,
                              unsigned* __restrict__ countAbove) {
    int b = threadIdx.x;
    if (b >= BATCH) return;
    const unsigned* h = hist + (size_t)b * NB;
    unsigned cum = 0, above = 0;
    int tb = 0;
    bool found = false;
    for (int j = NB - 1; j >= 0; --j) {
        unsigned c = h[j];
        if (cum + c >= KDET) { tb = j; above = cum; found = true; break; }
        cum += c;
    }
    if (!found) { tb = 0; above = cum; }
    threshBucket[b] = (unsigned)tb;
    countAbove[b]   = above;
}

// ---------------- kernel 3: collect candidates >= bucket floor --------------
// Second pass over cls_scores: L2-resident after pass 1 (128MB < 192MB L2).
__global__ __launch_bounds__(256) void collect_kernel(const float* __restrict__ cls,
                                                      const unsigned* __restrict__ threshBucket,
                                                      unsigned* __restrict__ candCount,
                                                      unsigned* __restrict__ candKey,
                                                      unsigned* __restrict__ candIdx) {
    const int b   = blockIdx.x / PB;
    const int blk = blockIdx.x % PB;
    const unsigned floorKey = threshBucket[b] << 20;
    const float4* base = reinterpret_cast<const float4*>(cls + (size_t)b * NELEM);
    const int nvec = NELEM / 4;
    const int stride = PB * 256;
    unsigned* ck = candKey + (size_t)b * CAND_MAX;
    unsigned* ci = candIdx + (size_t)b * CAND_MAX;
    for (int i = blk * 256 + threadIdx.x; i < nvec; i += stride) {
        __builtin_prefetch(base + i + stride, 0, 0);   // global_prefetch_b8
        float4 v = base[i];
        float e[4] = {v.x, v.y, v.z, v.w};
        #pragma unroll
        for (int j = 0; j < 4; ++j) {
            unsigned key = mono_key(e[j]);
            if (key >= floorKey) {
                unsigned pos = atomicAdd(&candCount[b], 1u);
                if (pos < CAND_MAX) { ck[pos] = key; ci[pos] = (unsigned)(i * 4 + j); }
            }
        }
    }
}

// ---------------- kernel 4: sort candidates, emit top-300 + decode ----------
__global__ __launch_bounds__(1024) void final_kernel(const float* __restrict__ bbox,
                                                     const unsigned* __restrict__ candCount,
                                                     const unsigned* __restrict__ candKey,
                                                     const unsigned* __restrict__ candIdx,
                                                     float* __restrict__ out) {
    __shared__ unsigned sKey[CAND_MAX];  // 32 KB
    __shared__ unsigned sIdx[CAND_MAX];  // 32 KB  (fits easily in 320KB WGP LDS)
    const int b = blockIdx.x;
    const int tid = threadIdx.x;
    unsigned cnt = candCount[b];
    if (cnt > CAND_MAX) cnt = CAND_MAX;
    const unsigned* ck = candKey + (size_t)b * CAND_MAX;
    const unsigned* ci = candIdx + (size_t)b * CAND_MAX;
    for (int i = tid; i < CAND_MAX; i += 1024) {
        if (i < (int)cnt) { sKey[i] = ck[i]; sIdx[i] = ci[i]; }
        else              { sKey[i] = 0u;    sIdx[i] = 0u;    }
    }
    __syncthreads();

    // bitonic sort, descending by key
    for (int k2 = 2; k2 <= CAND_MAX; k2 <<= 1) {
        for (int j = k2 >> 1; j > 0; j >>= 1) {
            for (int i = tid; i < CAND_MAX; i += 1024) {
                int ixj = i ^ j;
                if (ixj > i) {
                    bool desc = ((i & k2) == 0);
                    unsigned a = sKey[i], c = sKey[ixj];
                    if (desc ? (a < c) : (a > c)) {
                        sKey[i] = c; sKey[ixj] = a;
                        unsigned t = sIdx[i]; sIdx[i] = sIdx[ixj]; sIdx[ixj] = t;
                    }
                }
            }
            __syncthreads();
        }
    }

    if (tid < KDET) {
        unsigned key = sKey[tid];
        unsigned idx = sIdx[tid];
        float x = key_to_float(key);
        float score = 1.0f / (1.0f + expf(-x));        // sigmoid of logit
        int label = (int)(idx % CN);
        int query = (int)(idx / CN);
        const float* r = bbox + ((size_t)b * QN + query) * CN;
        float* bo = out + ((size_t)b * KDET + tid) * 10;
        bo[0] = r[0]; bo[1] = r[1]; bo[2] = r[2];       // x,y,z
        bo[3] = r[3]; bo[4] = r[4]; bo[5] = r[5];       // w,l,h
        bo[6] = atan2f(r[6], r[7]);                     // yaw
        bo[7] = r[8]; bo[8] = r[9]; bo[9] = 0.0f;       // vx,vy,vz
        float* scores = out + (size_t)BATCH * KDET * 10;
        float* labels = scores + (size_t)BATCH * KDET;
        scores[(size_t)b * KDET + tid] = score;
        labels[(size_t)b * KDET + tid] = (float)label;
    }
}

// ---------------- host launcher ---------------------------------------------
extern "C" void kernel_launch(void* const* d_in, const int* in_sizes, int n_in,
                              void* d_out, int out_size, void* d_ws, size_t ws_size,
                              hipStream_t stream) {
    (void)in_sizes; (void)n_in; (void)out_size; (void)ws_size;
    const float* cls  = (const float*)d_in[0];
    const float* bbox = (const float*)d_in[1];
    float* out = (float*)d_out;

    // workspace layout (u32 units), ~2.6 MB total
    unsigned* hist         = (unsigned*)d_ws;                       // BATCH*NB
    unsigned* candCount    = hist + (size_t)BATCH * NB;             // BATCH
    unsigned* threshBucket = candCount + BATCH;                     // BATCH
    unsigned* countAbove   = threshBucket + BATCH;                  // BATCH
    unsigned* candKey      = countAbove + BATCH;                    // BATCH*CAND_MAX
    unsigned* candIdx      = candKey + (size_t)BATCH * CAND_MAX;    // BATCH*CAND_MAX

    const int nZero = BATCH * NB + 3 * BATCH;   // hist + candCount + thresh + above
    init_kernel<<<(nZero + 255) / 256, 256, 0, stream>>>(hist, nZero);
    hist_kernel<<<BATCH * PB, 256, 0, stream>>>(cls, hist);
    thresh_kernel<<<1, BATCH, 0, stream>>>(hist, threshBucket, countAbove);
    collect_kernel<<<BATCH * PB, 256, 0, stream>>>(cls, threshBucket, candCount, candKey, candIdx);
    final_kernel<<<BATCH, 1024, 0, stream>>>(bbox, candCount, candKey, candIdx, out);
}